// Model_53291954209026
// MI455X (gfx1250) — compile-verified
//
#include <hip/hip_runtime.h>

// ---- problem constants (match reference) ------------------------------------
#define NB_TASKS        20
#define CLASSES         100
#define NCOLS           2000          // NB_TASKS * CLASSES
#define ROWS_PER_BLOCK  8             // 8 wave32 per 256-thread block
#define TEMP_INV        0.2f          // 1 / TEMPERATURE(5.0)

typedef __attribute__((ext_vector_type(4))) float v4f;
typedef int v4i_vs __attribute__((vector_size(16)));   // matches builtin param

#if defined(__gfx1250__) && __has_builtin(__builtin_amdgcn_global_load_async_to_lds_b128)
#define USE_ASYNC_LDS 1
#endif

// ---- wave32 cross-lane reductions -------------------------------------------
__device__ __forceinline__ float wave_max_f32(float v) {
#pragma unroll
  for (int off = 16; off > 0; off >>= 1)
    v = fmaxf(v, __shfl_xor(v, off, 32));
  return v;
}

__device__ __forceinline__ float wave_sum_f32(float v) {
#pragma unroll
  for (int off = 16; off > 0; off >>= 1)
    v += __shfl_xor(v, off, 32);
  return v;
}

// Per-lane gate value v (lanes 0..NB_TASKS-1 valid, others -inf).
// Returns hard_top2(v) * softmax(v / T) for this lane (0 for inactive lanes).
__device__ __forceinline__ float masked_softmax_weight(float v, int lane) {
  const bool active = (lane < NB_TASKS);
  const float m1 = wave_max_f32(v);
  // knock out one instance of the max, re-max -> second-largest value
  unsigned long long eq = __ballot(v == m1);
  const int first = __ffsll(eq) - 1;
  const float v2 = (lane == first) ? -__builtin_inff() : v;
  const float m2 = wave_max_f32(v2);
  const float hard = (active && (v >= m2)) ? 1.0f : 0.0f;
  const float e = active ? __expf((v - m1) * TEMP_INV) : 0.0f;
  const float s = wave_sum_f32(e);
  return hard * (e / s);
}

// ---- main fused kernel: one wave32 per batch row ----------------------------
__global__ __launch_bounds__(ROWS_PER_BLOCK * 32)
void task_gate_mix_kernel(const float* __restrict__ gol,      // [B, 20]
                          const float* __restrict__ gocls,    // [B, 2000]
                          const float* __restrict__ do_logit, // [B, 2000]
                          const float* __restrict__ alpha,    // [1]
                          float* __restrict__ out,            // [B, 2000]
                          int nrows) {
  __shared__ float s_gocls[ROWS_PER_BLOCK * NCOLS];   // 62.5 KB staging
  __shared__ float s_cw[ROWS_PER_BLOCK * NB_TASKS];   // fused per-task weights

  const int wv   = threadIdx.x >> 5;
  const int lane = threadIdx.x & 31;
  const int row  = blockIdx.x * ROWS_PER_BLOCK + wv;
  if (row >= nrows) return;                            // wave-uniform exit

  const float* __restrict__ gsrc = gocls    + (size_t)row * NCOLS;
  const float* __restrict__ dsrc = do_logit + (size_t)row * NCOLS;
  float*       __restrict__ ldst = &s_gocls[wv * NCOLS];

  // --- 1) kick off async gocls row -> LDS (ASYNCcnt-tracked DMA path) -------
#if defined(USE_ASYNC_LDS)
#pragma unroll 4
  for (int j = lane; j < NCOLS / 4; j += 32) {
    __builtin_amdgcn_global_load_async_to_lds_b128(
        (v4i_vs*)(gsrc + 4 * j),
        (v4i_vs*)(ldst + 4 * j),
        0, 0);
  }
#else
#pragma unroll 4
  for (int j = lane; j < NCOLS / 4; j += 32)
    ((v4f*)ldst)[j] = ((const v4f*)gsrc)[j];
#endif

  // --- 2) prefetch the do_logit row while the copy is in flight --------------
  __builtin_prefetch(dsrc + lane * 32, 0, 0);          // covers first 4 KB
  __builtin_prefetch(dsrc + 1024 + lane * 32, 0, 0);   // covers second 4 KB

  // --- 3) gol gate: softmax/T + top-2 over 20 tasks (wave32 reductions) ------
  const float a  = alpha[0];
  const float a1 = 1.0f - a;
  float gv_gol = (lane < NB_TASKS) ? gol[(size_t)row * NB_TASKS + lane]
                                   : -__builtin_inff();
  const float w_mcm = masked_softmax_weight(gv_gol, lane);

  // --- 4) wait for async staging, then per-lane segmented max ----------------
#if defined(USE_ASYNC_LDS)
#if __has_builtin(__builtin_amdgcn_s_wait_asynccnt)
  __builtin_amdgcn_s_wait_asynccnt(0);
#else
  asm volatile("s_wait_asynccnt 0" ::: "memory");
#endif
  asm volatile("" ::: "memory");                       // keep LDS reads below
#endif

  // Lane t owns task t: 25 x ds_load_b128 over its contiguous 100 floats.
  // Result lands directly in the lane that the gate reduction expects --
  // no cross-lane shuffles needed for the segmented max.
  float gv_cls = -__builtin_inff();
  if (lane < NB_TASKS) {
    const v4f* __restrict__ trow = (const v4f*)(ldst + lane * CLASSES);
    v4f m4 = trow[0];
#pragma unroll
    for (int k = 1; k < CLASSES / 4; ++k) {
      const v4f d = trow[k];
      m4.x = fmaxf(m4.x, d.x);
      m4.y = fmaxf(m4.y, d.y);
      m4.z = fmaxf(m4.z, d.z);
      m4.w = fmaxf(m4.w, d.w);
    }
    gv_cls = fmaxf(fmaxf(m4.x, m4.y), fmaxf(m4.z, m4.w));
  }
  const float w_gcl = masked_softmax_weight(gv_cls, lane);

  // --- 5) fuse both gates into one per-task scale ---------------------------
  if (lane < NB_TASKS)
    s_cw[wv * NB_TASKS + lane] = a * w_mcm + a1 * w_gcl;
  // wave-private LDS slice: no barrier needed, DScnt wait handled by compiler

  // --- 6) streaming pass: out = do_logit * cw[task], 128-bit NT lds/sts -----
  const v4f* __restrict__ d4 = (const v4f*)dsrc;
  v4f* __restrict__       o4 = (v4f*)(out + (size_t)row * NCOLS);
  const float* __restrict__ cw = &s_cw[wv * NB_TASKS];
#pragma unroll 4
  for (int j = lane; j < NCOLS / 4; j += 32) {
    v4f d = __builtin_nontemporal_load(d4 + j);
    const float w = cw[j / 25];                        // task = (4j)/100 = j/25
    __builtin_nontemporal_store(d * w, o4 + j);
  }
}

// ---- entry point ------------------------------------------------------------
extern "C" void kernel_launch(void* const* d_in, const int* in_sizes, int n_in,
                              void* d_out, int out_size, void* d_ws, size_t ws_size,
                              hipStream_t stream) {
  (void)n_in; (void)out_size; (void)d_ws; (void)ws_size;
  const float* gol      = (const float*)d_in[0];
  const float* gocls    = (const float*)d_in[1];
  const float* do_logit = (const float*)d_in[2];
  const float* alpha    = (const float*)d_in[3];
  float*       out      = (float*)d_out;

  const int nrows = in_sizes[0] / NB_TASKS;            // 8192
  dim3 grid((nrows + ROWS_PER_BLOCK - 1) / ROWS_PER_BLOCK);
  dim3 block(ROWS_PER_BLOCK * 32);
  task_gate_mix_kernel<<<grid, block, 0, stream>>>(gol, gocls, do_logit, alpha,
                                                   out, nrows);
}